// CPD_12498354831804
// MI455X (gfx1250) — compile-verified
//
#include <hip/hip_runtime.h>
#include <math.h>

#define N_DATA 500000
#define EMB 256
#define KNEG 4096
#define BS 64
#define NROW (KNEG + 1)          // 4097 rows per batch
#define NTILE ((NROW + 15) / 16) // 257 tiles of 16 rows
#define INV_T (1.0f / 0.07f)

typedef __attribute__((ext_vector_type(2))) float v2f;
typedef __attribute__((ext_vector_type(4))) float v4f;
typedef __attribute__((ext_vector_type(8))) float v8f;

// ---------------------------------------------------------------------------
// Kernel 1: L2-normalize feature rows.  grid=(64,2), block=256
// z=0 -> vis_feat -> vn_ws ; z=1 -> text_feat -> tn_ws
// ---------------------------------------------------------------------------
__global__ __launch_bounds__(256) void cpd_normalize_feat(
    const float* __restrict__ vis_feat, const float* __restrict__ text_feat,
    float* __restrict__ vn_ws, float* __restrict__ tn_ws) {
  __shared__ float red[256];
  const int b = blockIdx.x, z = blockIdx.y, t = threadIdx.x;
  const float* src = (z == 0 ? vis_feat : text_feat) + b * EMB;
  float v = src[t];
  red[t] = v * v;
  __syncthreads();
  for (int off = 128; off > 0; off >>= 1) {
    if (t < off) red[t] += red[t + off];
    __syncthreads();
  }
  float inv = 1.0f / fmaxf(sqrtf(red[0]), 1e-12f);
  float* dst = (z == 0 ? vn_ws : tn_ws) + b * EMB;
  dst[t] = v * inv;
}

// ---------------------------------------------------------------------------
// Kernel 2: WMMA gather mat-vec + exp.  grid=(257,64), block=32 (one wave).
// One wave per 16-row tile of one batch.  A = gathered 16x4 f32 tile,
// B = feature vector broadcast into all 16 columns, accumulated with
// v_wmma_f32_16x16x4_f32 over K=256.  All D columns are identical; lanes
// 0 and 16 (N=0) hold rows 0-7 / 8-15 in c[0..7].
// ---------------------------------------------------------------------------
__global__ __launch_bounds__(32) void cpd_score_wmma(
    const float* __restrict__ memBank,   // bank being gathered [N_DATA*EMB]
    const float* __restrict__ featN,     // normalized features [BS*EMB]
    const int* __restrict__ slct,        // [BS*NROW]
    float* __restrict__ out,             // exp scores [BS*NROW]
    float* __restrict__ partials)        // [BS*NTILE] per-tile sums
{
  __shared__ __align__(16) float f_lds[EMB];
  const int tile = blockIdx.x;
  const int b = blockIdx.y;
  const int lane = threadIdx.x;

  // Stage the 256-float feature row into LDS (8 floats per lane).
  const v4f* frow = (const v4f*)(featN + b * EMB);
  ((v4f*)f_lds)[lane]      = frow[lane];
  ((v4f*)f_lds)[lane + 32] = frow[lane + 32];
  __syncthreads();

  // A-matrix addressing: lane L<16 holds row M=L, K pair {k0,k0+1};
  // lane L+16 holds row M=L, K pair {k0+2,k0+3}.
  const int m = lane & 15;
  const int khalf = (lane >> 4) * 2;
  int row = tile * 16 + m;
  int srow = row < NROW ? row : NROW - 1;     // clamp tail tile (EXEC stays full)
  const long long g = (long long)slct[b * NROW + srow];
  const float* arow = memBank + g * EMB + khalf;
  const float* bvec = f_lds + khalf;          // same K slice; identical all columns

  v8f c0 = {}, c1 = {};
#pragma unroll 8
  for (int k0 = 0; k0 < EMB; k0 += 8) {
    v2f a0 = *(const v2f*)(arow + k0);
    v2f b0 = *(const v2f*)(bvec + k0);
    c0 = __builtin_amdgcn_wmma_f32_16x16x4_f32(false, a0, false, b0,
                                               (short)0, c0, false, false);
    v2f a1 = *(const v2f*)(arow + k0 + 4);
    v2f b1 = *(const v2f*)(bvec + k0 + 4);
    c1 = __builtin_amdgcn_wmma_f32_16x16x4_f32(false, a1, false, b1,
                                               (short)0, c1, false, false);
  }
  v8f c = c0 + c1;

  // lanes 0 and 16 hold all 16 dot products (column N=0).
  float s = 0.0f;
  if ((lane & 15) == 0) {
    const int mbase = tile * 16 + (lane >> 4) * 8;
#pragma unroll
    for (int r = 0; r < 8; ++r) {
      int orow = mbase + r;
      if (orow < NROW) {
        float e = expf(c[r] * INV_T);
        out[b * NROW + orow] = e;
        s += e;
      }
    }
  }
  s += __shfl_xor(s, 16, 32);
  if (lane == 0) partials[b * NTILE + tile] = s;
}

// ---------------------------------------------------------------------------
// Kernel 3: deterministic reduce of per-tile partials.  grid=2, block=256.
// ---------------------------------------------------------------------------
__global__ __launch_bounds__(256) void cpd_reduce(const float* __restrict__ partials,
                                                  float* __restrict__ sums) {
  __shared__ float red[256];
  const int n = BS * NTILE;
  const float* p = partials + blockIdx.x * n;
  float s = 0.0f;
  for (int i = threadIdx.x; i < n; i += 256) s += p[i];
  red[threadIdx.x] = s;
  __syncthreads();
  for (int off = 128; off > 0; off >>= 1) {
    if (threadIdx.x < off) red[threadIdx.x] += red[threadIdx.x + off];
    __syncthreads();
  }
  if (threadIdx.x == 0) sums[blockIdx.x] = red[0];
}

// ---------------------------------------------------------------------------
// Kernel 4: scale scores by BS*NROW / (sum * N_DATA)   (== divide by Z)
// ---------------------------------------------------------------------------
__global__ __launch_bounds__(256) void cpd_scale(float* __restrict__ out_vis,
                                                 float* __restrict__ out_text,
                                                 const float* __restrict__ sums) {
  const int total = BS * NROW;
  int i = blockIdx.x * blockDim.x + threadIdx.x;
  if (i < total) {
    out_vis[i]  *= (float)total / (sums[0] * (float)N_DATA);
    out_text[i] *= (float)total / (sums[1] * (float)N_DATA);
  }
}

// ---------------------------------------------------------------------------
// Kernel 5: streaming full-bank copy (native float4 vector, nontemporal
// load/store so the 1GB stream doesn't pollute L2)
// ---------------------------------------------------------------------------
__global__ __launch_bounds__(256) void cpd_copy4(const v4f* __restrict__ src,
                                                 v4f* __restrict__ dst,
                                                 long long n4) {
  long long i = (long long)blockIdx.x * blockDim.x + threadIdx.x;
  if (i < n4) {
    v4f v = __builtin_nontemporal_load(&src[i]);
    __builtin_nontemporal_store(v, &dst[i]);
  }
}

// ---------------------------------------------------------------------------
// Kernel 6: EMA update + renormalize the 64 touched rows.  grid=64, block=256.
// ---------------------------------------------------------------------------
__global__ __launch_bounds__(256) void cpd_update_rows(
    const float* __restrict__ memBank, const float* __restrict__ featN,
    const int* __restrict__ idx, float* __restrict__ outMem) {
  __shared__ float red[256];
  const int b = blockIdx.x, t = threadIdx.x;
  const long long row = (long long)idx[b];
  float pos = memBank[row * EMB + t] * 0.5f + featN[b * EMB + t] * 0.5f;
  red[t] = pos * pos;
  __syncthreads();
  for (int off = 128; off > 0; off >>= 1) {
    if (t < off) red[t] += red[t + off];
    __syncthreads();
  }
  float inv = 1.0f / fmaxf(sqrtf(red[0]), 1e-12f);
  outMem[row * EMB + t] = pos * inv;
}

extern "C" void kernel_launch(void* const* d_in, const int* in_sizes, int n_in,
                              void* d_out, int out_size, void* d_ws, size_t ws_size,
                              hipStream_t stream) {
  const float* vis_feat    = (const float*)d_in[0];
  const float* text_feat   = (const float*)d_in[1];
  const float* vis_memory  = (const float*)d_in[2];
  const float* text_memory = (const float*)d_in[3];
  const int*   slct_idx    = (const int*)d_in[4];
  const int*   idx         = (const int*)d_in[5];

  float* out_vis  = (float*)d_out;                 // [BS*NROW]
  float* out_text = out_vis + (size_t)BS * NROW;   // [BS*NROW]
  float* out_vm   = out_text + (size_t)BS * NROW;  // [N_DATA*EMB]
  float* out_tm   = out_vm + (size_t)N_DATA * EMB; // [N_DATA*EMB]

  // Workspace layout (floats): [0..1]=sums, vn[BS*EMB], tn[BS*EMB],
  // partials_vis[BS*NTILE], partials_text[BS*NTILE]
  float* wsf      = (float*)d_ws;
  float* sums     = wsf;
  float* vn_ws    = wsf + 8;
  float* tn_ws    = vn_ws + BS * EMB;
  float* partials = tn_ws + BS * EMB;              // vis then text, contiguous

  // 1) normalize features
  cpd_normalize_feat<<<dim3(BS, 2), 256, 0, stream>>>(vis_feat, text_feat,
                                                      vn_ws, tn_ws);
  // 2) WMMA gather mat-vec + exp:  vis_out uses text bank & vn; text_out vice versa
  cpd_score_wmma<<<dim3(NTILE, BS), 32, 0, stream>>>(
      text_memory, vn_ws, slct_idx, out_vis, partials);
  cpd_score_wmma<<<dim3(NTILE, BS), 32, 0, stream>>>(
      vis_memory, tn_ws, slct_idx, out_text, partials + BS * NTILE);
  // 3) deterministic global sums
  cpd_reduce<<<2, 256, 0, stream>>>(partials, sums);
  // 4) divide by Z
  cpd_scale<<<(BS * NROW + 255) / 256, 256, 0, stream>>>(out_vis, out_text, sums);
  // 5) stream the two 512MB banks to the output
  const long long n4 = (long long)N_DATA * EMB / 4;
  const int cpblocks = (int)((n4 + 255) / 256);
  cpd_copy4<<<cpblocks, 256, 0, stream>>>((const v4f*)vis_memory,
                                          (v4f*)out_vm, n4);
  cpd_copy4<<<cpblocks, 256, 0, stream>>>((const v4f*)text_memory,
                                          (v4f*)out_tm, n4);
  // 6) EMA-update + renormalize touched rows (after copies in stream order)
  cpd_update_rows<<<BS, 256, 0, stream>>>(vis_memory, vn_ws, idx, out_vm);
  cpd_update_rows<<<BS, 256, 0, stream>>>(text_memory, tn_ws, idx, out_tm);
}